// SOAP_predictor_model_56727928046235
// MI455X (gfx1250) — compile-verified
//
#include <hip/hip_runtime.h>
#include <math.h>

// ---------------------------------------------------------------------------
// SOAP predictor for MI455X (gfx1250).  fp32 end-to-end; matrix work lowered
// to V_WMMA_F32_16X16X4_F32 (wave32).  Compute-bound (<1MB data, L2-resident).
// Round 3: fast-rcp sigmoid (v_rcp_f32 instead of IEEE div_scale chains) in
// the gated epilogues; otherwise identical to round 2 (per-wave column-tile
// ownership, 8 WMMAs per B fragment, WMMA NCE aggregation, logits trick).
// ---------------------------------------------------------------------------

typedef float v2f __attribute__((ext_vector_type(2)));
typedef float v8f __attribute__((ext_vector_type(8)));

#if defined(__gfx1250__) && __has_builtin(__builtin_amdgcn_wmma_f32_16x16x4_f32)
#define HAVE_WMMA_F32 1
#else
#define HAVE_WMMA_F32 0
#endif

#define DEVINL __device__ __forceinline__

DEVINL float fastrcp_(float x) {
#if defined(__AMDGCN__) && __has_builtin(__builtin_amdgcn_rcpf)
    return __builtin_amdgcn_rcpf(x);   // v_rcp_f32, ~1 ulp
#else
    return 1.f / x;
#endif
}
DEVINL float sigm_(float x) { return fastrcp_(1.f + __expf(-x)); }

struct KP {
    const float *positions, *node_extra;
    const float *sa_wq0, *sa_wk0, *sa_wv0;
    const float *sa_wq1, *sa_wk1, *sa_wv1;
    const float *sa_wq2, *sa_wk2, *sa_wv2;
    const float *sa_wq3, *sa_wk3, *sa_wv3;
    const float *nce_wg, *nce_wm;
    const float *ge_w1, *ge_w2, *ge_wp;
    const float *gru_wi, *gru_wh, *gru_bi, *gru_bh;
    const float *ag_wg, *ag_wm;
    const float *gc_wi, *gc_wh, *gc_bi, *gc_bh;
    const float *hd_w0, *hd_b0, *hd_g0, *hd_m0;
    const float *hd_w1, *hd_b1, *hd_g1, *hd_m1;
    const float *hd_w2, *hd_b2, *hd_g2, *hd_m2;
    const float *hd_w3, *hd_b3, *hd_g3, *hd_m3;
    const int   *species;
    float *feat6, *x13, *emask, *nce, *hG, *gcen, *out;
};

// ===========================================================================
// Shared WMMA workhorse: given h [64 x K] in LDS (row stride 128) and global
// row-major Wg, Wm [K x N], overwrite h in place with sigmoid(h@Wg)*(h@Wm).
// Wave w owns column tile ct = w % NCT and RPW row tiles; B fragments are
// loaded once per k-step and reused across all row tiles (and both chains
// share the A fragment).  All branching is wave-uniform -> EXEC all-ones.
// ===========================================================================
template <int K, int N>
DEVINL void gated_mm(float* sh, const float* __restrict__ Wg,
                     const float* __restrict__ Wm, int tid) {
#if HAVE_WMMA_F32
    constexpr int NCT = N / 16;                      // column tiles
    constexpr int RPW = (NCT >= 2) ? (NCT / 2) : 1;  // row tiles per wave
    int wave = tid >> 5, lane = tid & 31;
    int mn = lane & 15;            // row-in-tile (A) / col-in-tile (B)
    int kb = (lane >> 4) << 1;     // k-pair offset within 4-wide k tile
    int ct = wave % NCT;
    int rt0 = (wave / NCT) * RPW;
    bool active = rt0 < 4;
    v8f AG[RPW], AM[RPW];
    if (active) {
        v8f zero = {};
#pragma unroll
        for (int r = 0; r < RPW; ++r) { AG[r] = zero; AM[r] = zero; }
        const float* wgp = Wg + kb * N + ct * 16 + mn;
        const float* wmp = Wm + kb * N + ct * 16 + mn;
        const float* hbase = sh + mn * 128 + kb;
        for (int k0 = 0; k0 < K; k0 += 4) {
            v2f bg; bg.x = wgp[k0 * N]; bg.y = wgp[k0 * N + N];
            v2f bm; bm.x = wmp[k0 * N]; bm.y = wmp[k0 * N + N];
#pragma unroll
            for (int r = 0; r < RPW; ++r) {
                const float* hrow = hbase + (rt0 + r) * (16 * 128);
                v2f a; a.x = hrow[k0]; a.y = hrow[k0 + 1];
                AG[r] = __builtin_amdgcn_wmma_f32_16x16x4_f32(false, a, false, bg,
                                                              (short)0, AG[r], false, false);
                AM[r] = __builtin_amdgcn_wmma_f32_16x16x4_f32(false, a, false, bm,
                                                              (short)0, AM[r], false, false);
            }
        }
    }
    __syncthreads();  // all reads of h complete before in-place overwrite
    if (active) {
#pragma unroll
        for (int r = 0; r < RPW; ++r) {
            int col = ct * 16 + mn;
            int rbase = (rt0 + r) * 16 + ((lane >> 4) << 3);
#pragma unroll
            for (int j = 0; j < 8; ++j)
                sh[(rbase + j) * 128 + col] = AM[r][j] * sigm_(AG[r][j]);
        }
    }
#else
    // scalar fallback: two-phase in-place update via register staging
    constexpr int PER = (64 * N + 255) / 256;
    float tmp[PER];
    int c2 = 0;
    for (int idx = tid; idx < 64 * N; idx += 256) {
        int row = idx / N, col = idx - row * N;
        const float* hr = sh + row * 128;
        float sg = 0.f, sm = 0.f;
        for (int k = 0; k < K; ++k) {
            float hv = hr[k];
            sg += hv * Wg[k * N + col];
            sm += hv * Wm[k * N + col];
        }
        tmp[c2++] = sm * sigm_(sg);
    }
    __syncthreads();
    c2 = 0;
    for (int idx = tid; idx < 64 * N; idx += 256) {
        int row = idx / N, col = idx - row * N;
        sh[row * 128 + col] = tmp[c2++];
    }
#endif
}

// ===========================================================================
// K0: per-node features, x13, env masks
// ===========================================================================
__global__ __launch_bounds__(64) void prep_kernel(KP p) {
    int b = blockIdx.x, m = threadIdx.x;
    int g = b * 64 + m;
    int sp = p.species[g];
    float f6[6];
    f6[0] = p.positions[g * 3 + 0];
    f6[1] = p.positions[g * 3 + 1];
    f6[2] = p.positions[g * 3 + 2];
    f6[3] = (sp == 0) ? 1.f : 0.f;
    f6[4] = (sp == 1) ? 1.f : 0.f;
    f6[5] = (sp == 2) ? 1.f : 0.f;
#pragma unroll
    for (int i = 0; i < 6; ++i) p.feat6[g * 6 + i] = f6[i];
#pragma unroll
    for (int i = 0; i < 6; ++i) p.x13[g * 13 + i] = f6[i];
#pragma unroll
    for (int i = 0; i < 7; ++i) p.x13[g * 13 + 6 + i] = p.node_extra[g * 7 + i];
    const int EL[6] = {0, 0, 0, 1, 1, 2};
    const int EH[6] = {0, 1, 2, 1, 2, 2};
#pragma unroll
    for (int n = 0; n < 6; ++n)
        p.emask[b * 384 + n * 64 + m] = (sp == EL[n] || sp == EH[n]) ? 1.f : 0.f;
}

// ===========================================================================
// K1: node-centre encoding.  One block per (molecule, centre), 256 threads.
// logits[m] = in[m].(Wk@q) (algebraic identity); v-matmul split over 4 column
// chunks with register-staged rows (safe in-place update).
// ===========================================================================
template <int DE, int DD>
DEVINL void attn_stage(const float* wq, const float* wk, const float* wv,
                       const float* cf, const float* ein, float* eout, int ldo,
                       const float* sf6, float* sq, float* swkq,
                       float* slog, float* sa_, int tid) {
    constexpr int DIN = DE + 6;
    constexpr int DC = DD / 4;   // outputs per (row, chunk) thread
    if (tid < DD) {
        float q = 0.f;
#pragma unroll
        for (int i = 0; i < 6; ++i) q += cf[i] * wq[i * DD + tid];
        sq[tid] = q;
    }
    __syncthreads();
    if (tid < DIN) {             // Wk @ q  (DIN x DD @ DD)
        float s = 0.f;
        for (int d = 0; d < DD; ++d) s += wk[tid * DD + d] * sq[d];
        swkq[tid] = s;
    }
    __syncthreads();
    if (tid < 64) {
        float lg = 0.f;
#pragma unroll
        for (int i = 0; i < DE; ++i) lg += ein[tid * 64 + i] * swkq[i];
#pragma unroll
        for (int i = 0; i < 6; ++i) lg += sf6[tid * 6 + i] * swkq[DE + i];
        slog[tid] = lg * rsqrtf((float)DD);
    }
    __syncthreads();
    if (tid == 0) {  // 64-wide softmax, serial (negligible cost)
        float mx = slog[0];
        for (int m = 1; m < 64; ++m) mx = fmaxf(mx, slog[m]);
        float sum = 0.f;
        for (int m = 0; m < 64; ++m) { float e = __expf(slog[m] - mx); sa_[m] = e; sum += e; }
        float inv = fastrcp_(sum);
        for (int m = 0; m < 64; ++m) sa_[m] *= inv;
    }
    __syncthreads();
    // v: thread = (row, chunk); stage the row in registers, then write in place
    int row = tid & 63, chunk = tid >> 6;
    float inr[DIN];
#pragma unroll
    for (int i = 0; i < DE; ++i) inr[i] = ein[row * 64 + i];
#pragma unroll
    for (int i = 0; i < 6; ++i) inr[DE + i] = sf6[row * 6 + i];
    float am = sa_[row];
    __syncthreads();  // all rows staged before in-place writes
    {
        int d0 = chunk * DC;
        for (int d = d0; d < d0 + DC; ++d) {
            float v = 0.f;
#pragma unroll
            for (int i = 0; i < DIN; ++i) v += inr[i] * wv[i * DD + d];
            eout[row * ldo + d] = am * v;
        }
    }
    __syncthreads();
}

__global__ __launch_bounds__(256) void nce_kernel(KP p) {
    __shared__ float sf6[64 * 6];
    __shared__ float se[64 * 64];     // stages 0..2 (widths 16/32/64)
    __shared__ float se3[64 * 128];   // stage 3 output / gated product
    __shared__ float sq[128];
    __shared__ float swkq[72];
    __shared__ float slog[64];
    __shared__ float sa_[64];
    int tid = threadIdx.x;
    int b = blockIdx.x >> 6, cC = blockIdx.x & 63;

    for (int idx = tid; idx < 384; idx += 256) sf6[idx] = p.feat6[b * 384 + idx];
    __syncthreads();
    float cf[6];
#pragma unroll
    for (int i = 0; i < 6; ++i) cf[i] = sf6[cC * 6 + i];

    attn_stage<0, 16 >(p.sa_wq0, p.sa_wk0, p.sa_wv0, cf, se, se, 64, sf6, sq, swkq, slog, sa_, tid);
    attn_stage<16, 32>(p.sa_wq1, p.sa_wk1, p.sa_wv1, cf, se, se, 64, sf6, sq, swkq, slog, sa_, tid);
    attn_stage<32, 64>(p.sa_wq2, p.sa_wk2, p.sa_wv2, cf, se, se, 64, sf6, sq, swkq, slog, sa_, tid);
    attn_stage<64, 128>(p.sa_wq3, p.sa_wk3, p.sa_wv3, cf, se, se3, 128, sf6, sq, swkq, slog, sa_, tid);

    // gated aggregation via WMMA: se3 <- sigmoid(e@wg)*(e@wm), then row-sum
    gated_mm<128, 128>(se3, p.nce_wg, p.nce_wm, tid);
    __syncthreads();
    if (tid < 128) {
        float acc = 0.f;
        for (int m = 0; m < 64; ++m) acc += se3[m * 128 + tid];
        p.nce[(b * 64 + cC) * 128 + tid] = acc;
    }
}

// ===========================================================================
// K2: graph encoding (GRU scan) + graph-centre GRU.  One block per molecule.
// ===========================================================================
__global__ __launch_bounds__(192) void graph_kernel(KP p) {
    __shared__ float sh0[64 * 16], sh48[64 * 48], shin[64 * 64], shsv[64 * 64];
    __shared__ float sh[64], sgi[192], sgh[192], sgate[64];
    __shared__ float ss[64], shG[128], sghc[384], sgic[384];
    int tid = threadIdx.x, b = blockIdx.x;
    const float* x13 = p.x13 + b * 832;

    for (int idx = tid; idx < 1024; idx += 192) {
        int m = idx >> 4, c = idx & 15;
        float s = 0.f;
#pragma unroll
        for (int i = 0; i < 13; ++i) s += x13[m * 13 + i] * p.ge_w1[i * 16 + c];
        sh0[idx] = tanhf(s);
    }
    __syncthreads();
    for (int idx = tid; idx < 3072; idx += 192) {
        int m = idx / 48, c = idx % 48;
        float s = 0.f;
#pragma unroll
        for (int i = 0; i < 16; ++i) s += sh0[m * 16 + i] * p.ge_w2[i * 48 + c];
        sh48[idx] = s;
    }
    __syncthreads();
    for (int idx = tid; idx < 4096; idx += 192) {
        int m = idx >> 6, c = idx & 63;
        float s = 0.f;
        for (int i = 0; i < 48; ++i) s += sh48[m * 48 + i] * p.ge_wp[i * 64 + c];
        shin[idx] = tanhf(s);
    }
    if (tid < 64) sh[tid] = 0.f;
    __syncthreads();

    for (int step = 0; step < 64; ++step) {
        {
            float gi = p.gru_bi[tid], gh = p.gru_bh[tid];
            const float* x = shin + step * 64;
            for (int i = 0; i < 64; ++i) {
                gi += x[i] * p.gru_wi[i * 192 + tid];
                gh += sh[i] * p.gru_wh[i * 192 + tid];
            }
            sgi[tid] = gi; sgh[tid] = gh;
        }
        __syncthreads();
        if (tid < 64) {
            float z = sigm_(sgi[tid] + sgh[tid]);
            float r = sigm_(sgi[64 + tid] + sgh[64 + tid]);
            float n = tanhf(sgi[128 + tid] + r * sgh[128 + tid]);
            sgate[tid] = (1.f - z) * n + z * sh[tid];
        }
        __syncthreads();
        if (tid < 64) { sh[tid] = sgate[tid]; shsv[step * 64 + tid] = sgate[tid]; }
        __syncthreads();
    }

    if (tid < 64) {
        float acc = 0.f;
        for (int m = 0; m < 64; ++m) {
            const float* hr = shsv + m * 64;
            float g = 0.f;
            for (int i = 0; i < 64; ++i) g += hr[i] * p.ag_wg[i * 64 + tid];
            acc += sigm_(g) * hr[tid];
        }
        ss[tid] = acc;
    }
    __syncthreads();
    if (tid < 128) {
        float s = 0.f;
        for (int i = 0; i < 64; ++i) s += ss[i] * p.ag_wm[i * 128 + tid];
        shG[tid] = s;
        p.hG[b * 128 + tid] = s;
    }
    __syncthreads();

    for (int t = tid; t < 384; t += 192) {
        float s = p.gc_bh[t];
        for (int i = 0; i < 128; ++i) s += shG[i] * p.gc_wh[i * 384 + t];
        sghc[t] = s;
    }
    __syncthreads();
    for (int m = 0; m < 64; ++m) {
        const float* x = p.nce + (b * 64 + m) * 128;
        for (int t = tid; t < 384; t += 192) {
            float s = p.gc_bi[t];
            for (int i = 0; i < 128; ++i) s += x[i] * p.gc_wi[i * 384 + t];
            sgic[t] = s;
        }
        __syncthreads();
        if (tid < 128) {
            float z = sigm_(sgic[tid] + sghc[tid]);
            float r = sigm_(sgic[128 + tid] + sghc[128 + tid]);
            float n = tanhf(sgic[256 + tid] + r * sghc[256 + tid]);
            p.gcen[(b * 64 + m) * 128 + tid] = (1.f - z) * n + z * shG[tid];
        }
        __syncthreads();
    }
}

// ===========================================================================
// K3: SOAP decoder heads.  One block per (molecule, atom centre), 256 threads.
// Layer-0 h is shared across all 6 envs (masks only affect the reduction).
// ===========================================================================
DEVINL void masked_reduce(const float* sh, const float* smask, float* sca,
                          int N, int e0, int e1, int tid) {
    int ne = e1 - e0 + 1;
    for (int idx = tid; idx < ne * N; idx += 256) {
        int n = e0 + idx / N, col = idx - (idx / N) * N;
        const float* mk = smask + n * 64;
        float s = 0.f;
        for (int row = 0; row < 64; ++row) s += mk[row] * sh[row * 128 + col];
        sca[n * 128 + col] = s;
    }
}

// h[row,col] = tanh(ctx@W_top[col] + rel[row]@W_bot[col] + b[col])
template <int KC, int N>
DEVINL void build_h(const float* ctx, const float* w, const float* bb,
                    float* sh, float* sctx, const float* srel, int tid) {
    if (tid < N) {
        float s = bb[tid];
        for (int i = 0; i < KC; ++i) s += ctx[i] * w[i * N + tid];
        sctx[tid] = s;
    }
    __syncthreads();
    for (int idx = tid; idx < 64 * N; idx += 256) {
        int r = idx / N, c = idx - (idx / N) * N;
        float s = sctx[c];
        const float* rl = srel + r * 6;
#pragma unroll
        for (int i = 0; i < 6; ++i) s += rl[i] * w[(KC + i) * N + c];
        sh[r * 128 + c] = tanhf(s);
    }
    __syncthreads();
}

__global__ __launch_bounds__(256) void head_kernel(KP p) {
    __shared__ float sh[64 * 128];   // h, then overwritten by sigmoid(hg)*hm
    __shared__ float scp[6 * 128];   // ctx per env (previous layer)
    __shared__ float sca[6 * 128];   // ctx per env (current layer)
    __shared__ float srel[64 * 6];
    __shared__ float smask[6 * 64];
    __shared__ float sctx[128];
    __shared__ float sgc[128];
    int tid = threadIdx.x;
    int b = blockIdx.x >> 6, mC = blockIdx.x & 63;

    if (tid < 128) sgc[tid] = p.gcen[(b * 64 + mC) * 128 + tid];
    for (int idx = tid; idx < 384; idx += 256) {
        int m = idx / 6, i = idx - (idx / 6) * 6;
        srel[idx] = (i < 3)
            ? p.positions[(b * 64 + m) * 3 + i] - p.positions[(b * 64 + mC) * 3 + i]
            : p.feat6[(b * 64 + m) * 6 + i];
    }
    for (int idx = tid; idx < 384; idx += 256) smask[idx] = p.emask[b * 384 + idx];
    __syncthreads();

    for (int l = 0; l < 4; ++l) {
        const float* w0 = p.hd_w0 + l * 134 * 128; const float* b0 = p.hd_b0 + l * 128;
        const float* g0 = p.hd_g0 + l * 128 * 128; const float* m0 = p.hd_m0 + l * 128 * 128;
        const float* w1 = p.hd_w1 + l * 134 * 64;  const float* b1 = p.hd_b1 + l * 64;
        const float* g1 = p.hd_g1 + l * 64 * 64;   const float* m1 = p.hd_m1 + l * 64 * 64;
        const float* w2 = p.hd_w2 + l * 70 * 32;   const float* b2 = p.hd_b2 + l * 32;
        const float* g2 = p.hd_g2 + l * 32 * 32;   const float* m2 = p.hd_m2 + l * 32 * 32;
        const float* w3 = p.hd_w3 + l * 38 * 16;   const float* b3 = p.hd_b3 + l * 16;
        const float* g3 = p.hd_g3 + l * 16 * 16;   const float* m3 = p.hd_m3 + l * 16 * 16;

        // ---- layer 0: h shared across all 6 envs
        build_h<128, 128>(sgc, w0, b0, sh, sctx, srel, tid);
        gated_mm<128, 128>(sh, g0, m0, tid);
        __syncthreads();
        masked_reduce(sh, smask, sca, 128, 0, 5, tid);
        __syncthreads();
        for (int idx = tid; idx < 768; idx += 256) scp[idx] = sca[idx];
        __syncthreads();

        // ---- layer 1: per env
        for (int n = 0; n < 6; ++n) {
            build_h<128, 64>(scp + n * 128, w1, b1, sh, sctx, srel, tid);
            gated_mm<64, 64>(sh, g1, m1, tid);
            __syncthreads();
            masked_reduce(sh, smask, sca, 64, n, n, tid);
            __syncthreads();
        }
        for (int idx = tid; idx < 768; idx += 256) scp[idx] = sca[idx];
        __syncthreads();

        // ---- layer 2
        for (int n = 0; n < 6; ++n) {
            build_h<64, 32>(scp + n * 128, w2, b2, sh, sctx, srel, tid);
            gated_mm<32, 32>(sh, g2, m2, tid);
            __syncthreads();
            masked_reduce(sh, smask, sca, 32, n, n, tid);
            __syncthreads();
        }
        for (int idx = tid; idx < 768; idx += 256) scp[idx] = sca[idx];
        __syncthreads();

        // ---- layer 3
        for (int n = 0; n < 6; ++n) {
            build_h<32, 16>(scp + n * 128, w3, b3, sh, sctx, srel, tid);
            gated_mm<16, 16>(sh, g3, m3, tid);
            __syncthreads();
            masked_reduce(sh, smask, sca, 16, n, n, tid);
            __syncthreads();
        }

        for (int idx = tid; idx < 96; idx += 256) {
            int n = idx >> 4, i2 = idx & 15;
            p.out[b * 24576 + (mC * 16 + i2) * 24 + n * 4 + l] = sca[n * 128 + i2];
        }
        __syncthreads();
    }
}

// ===========================================================================
extern "C" void kernel_launch(void* const* d_in, const int* in_sizes, int n_in,
                              void* d_out, int out_size, void* d_ws, size_t ws_size,
                              hipStream_t stream) {
    (void)in_sizes; (void)n_in; (void)out_size; (void)ws_size;
    KP p;
    p.positions = (const float*)d_in[0];  p.node_extra = (const float*)d_in[1];
    p.sa_wq0 = (const float*)d_in[2];  p.sa_wk0 = (const float*)d_in[3];  p.sa_wv0 = (const float*)d_in[4];
    p.sa_wq1 = (const float*)d_in[5];  p.sa_wk1 = (const float*)d_in[6];  p.sa_wv1 = (const float*)d_in[7];
    p.sa_wq2 = (const float*)d_in[8];  p.sa_wk2 = (const float*)d_in[9];  p.sa_wv2 = (const float*)d_in[10];
    p.sa_wq3 = (const float*)d_in[11]; p.sa_wk3 = (const float*)d_in[12]; p.sa_wv3 = (const float*)d_in[13];
    p.nce_wg = (const float*)d_in[14]; p.nce_wm = (const float*)d_in[15];
    p.ge_w1 = (const float*)d_in[16];  p.ge_w2 = (const float*)d_in[17];  p.ge_wp = (const float*)d_in[18];
    p.gru_wi = (const float*)d_in[19]; p.gru_wh = (const float*)d_in[20];
    p.gru_bi = (const float*)d_in[21]; p.gru_bh = (const float*)d_in[22];
    p.ag_wg = (const float*)d_in[23];  p.ag_wm = (const float*)d_in[24];
    p.gc_wi = (const float*)d_in[25];  p.gc_wh = (const float*)d_in[26];
    p.gc_bi = (const float*)d_in[27];  p.gc_bh = (const float*)d_in[28];
    p.hd_w0 = (const float*)d_in[29];  p.hd_b0 = (const float*)d_in[30];
    p.hd_g0 = (const float*)d_in[31];  p.hd_m0 = (const float*)d_in[32];
    p.hd_w1 = (const float*)d_in[33];  p.hd_b1 = (const float*)d_in[34];
    p.hd_g1 = (const float*)d_in[35];  p.hd_m1 = (const float*)d_in[36];
    p.hd_w2 = (const float*)d_in[37];  p.hd_b2 = (const float*)d_in[38];
    p.hd_g2 = (const float*)d_in[39];  p.hd_m2 = (const float*)d_in[40];
    p.hd_w3 = (const float*)d_in[41];  p.hd_b3 = (const float*)d_in[42];
    p.hd_g3 = (const float*)d_in[43];  p.hd_m3 = (const float*)d_in[44];
    p.species = (const int*)d_in[45];

    float* ws = (float*)d_ws;
    p.feat6 = ws;              // 8*64*6    = 3072
    p.x13   = ws + 3072;       // 8*64*13   = 6656
    p.emask = ws + 9728;       // 8*6*64    = 3072
    p.nce   = ws + 12800;      // 8*64*128  = 65536
    p.hG    = ws + 78336;      // 8*128     = 1024
    p.gcen  = ws + 79360;      // 8*64*128  = 65536  (ends 144896 floats)
    p.out = (float*)d_out;

    prep_kernel  <<<dim3(8),   dim3(64),  0, stream>>>(p);
    nce_kernel   <<<dim3(512), dim3(256), 0, stream>>>(p);
    graph_kernel <<<dim3(8),   dim3(192), 0, stream>>>(p);
    head_kernel  <<<dim3(512), dim3(256), 0, stream>>>(p);
}